// VectorQuantizerEMA_10170482556966
// MI455X (gfx1250) — compile-verified
//
#include <hip/hip_runtime.h>
#include <hip/hip_bf16.h>
#include <math.h>

#define EMB_D 64
#define NUM_K 2048
#define NUM_N 65536

typedef __attribute__((ext_vector_type(2))) float v2f;
typedef __attribute__((ext_vector_type(8))) float v8f;

// ---------------- workspace layout (bytes) ----------------
// [0)      s[2048]       float   (per-code sum(w_k))
// [8192)   counts[2048]  uint    (histogram of chosen codes)
// [16384)  sse           float   (sum of squared quantization error)
// [16640)  indices[65536] int
#define WS_S_OFF    0
#define WS_CNT_OFF  8192
#define WS_SSE_OFF  16384
#define WS_IDX_OFF  16640

// ---- 1) per-code row sums + zero accumulators (must re-zero every call) ----
__global__ void vq_prep(const float* __restrict__ w, float* __restrict__ s,
                        unsigned int* __restrict__ counts, float* __restrict__ sse) {
  int k = blockIdx.x * blockDim.x + threadIdx.x;   // 0..2047
  if (k < NUM_K) {
    const float* wr = w + k * EMB_D;
    float acc = 0.f;
#pragma unroll
    for (int j = 0; j < EMB_D; ++j) acc += wr[j];
    s[k] = acc;
    counts[k] = 0u;
  }
  if (k == 0) *sse = 0.f;
}

// ---- 2) zero the 512 MB one-hot region (8B-aligned -> float2 NT stores) ----
__global__ void vq_zerofill(float* __restrict__ enc) {
  v2f z = {};
  const size_t nPairs = (size_t)NUM_N * NUM_K / 2;
  const size_t stride = (size_t)gridDim.x * blockDim.x;
  for (size_t i = (size_t)blockIdx.x * blockDim.x + threadIdx.x; i < nPairs; i += stride) {
    __builtin_nontemporal_store(z, (v2f*)enc + i);
  }
}

// ---- 3) WMMA distance + argmin:
//   D[m,n] = sum_k w[m,k] * (-2*x[n,k]);  dist = D + s[m];  argmin over m.
//   A = codes (16xK f32), B = -2*x rows (Kx16 f32), 16x 16x16x4 WMMAs per tile,
//   split into two accumulator chains; A fragments double-buffered across tiles.
__global__ void __launch_bounds__(256)
vq_argmin(const float* __restrict__ x, const float* __restrict__ w,
          const float* __restrict__ s, unsigned int* __restrict__ counts,
          int* __restrict__ indices) {
  __shared__ float s_sh[NUM_K];
  __shared__ unsigned int hist[NUM_K];
  for (int i = threadIdx.x; i < NUM_K; i += 256) { s_sh[i] = s[i]; hist[i] = 0u; }
  __syncthreads();

  const int lane = threadIdx.x & 31;
  const int wv   = threadIdx.x >> 5;   // 8 waves / block
  const int h    = lane >> 4;          // K half: this lane holds K = 4*kk + 2h + {0,1}
  const int nl   = lane & 15;          // A row (code) within tile / B col (x row)
  const int row  = blockIdx.x * 128 + wv * 16 + nl;

  // Preload B fragments: 16 k-steps of (-2 * x[row]) in the f32 WMMA B layout.
  v2f bx[16];
  const float* xrow = x + ((size_t)row << 6) + (h << 1);
#pragma unroll
  for (int kk = 0; kk < 16; ++kk) {
    v2f t = *(const v2f*)(xrow + kk * 4);
    bx[kk] = t * -2.0f;
  }

  float bestV = 3.0e38f;
  int   bestI = 0;
  const float* wbase = w + ((size_t)nl << 6) + (h << 1);

  // Prime the A-fragment double buffer with tile 0.
  v2f aCur[16];
#pragma unroll
  for (int kk = 0; kk < 16; ++kk) aCur[kk] = *(const v2f*)(wbase + kk * 4);

#pragma unroll 2
  for (int t = 0; t < NUM_K / 16; ++t) {
    // Prefetch next tile's A fragments (branchless wrap keeps it in-bounds;
    // the wrapped loads on the final iteration are discarded).
    const float* wnext = wbase + ((size_t)((t + 1) & (NUM_K / 16 - 1)) << 10);
    v2f aNxt[16];
#pragma unroll
    for (int kk = 0; kk < 16; ++kk) aNxt[kk] = *(const v2f*)(wnext + kk * 4);

    // Two independent WMMA chains over the 16 K-steps.
    v8f acc0 = {}, acc1 = {};
#pragma unroll
    for (int kk = 0; kk < 8; ++kk) {
      acc0 = __builtin_amdgcn_wmma_f32_16x16x4_f32(false, aCur[2 * kk], false,
                                                   bx[2 * kk], (short)0, acc0,
                                                   false, false);
      acc1 = __builtin_amdgcn_wmma_f32_16x16x4_f32(false, aCur[2 * kk + 1], false,
                                                   bx[2 * kk + 1], (short)0, acc1,
                                                   false, false);
    }

    // Lane holds D for x-row nl, codes M = t*16 + 8h + v (v = VGPR 0..7).
    const int cbase = t * 16 + h * 8;
#pragma unroll
    for (int v = 0; v < 8; ++v) {
      float dv = (acc0[v] + acc1[v]) + s_sh[cbase + v];
      if (dv < bestV) { bestV = dv; bestI = cbase + v; }   // strict < keeps first min
    }

#pragma unroll
    for (int kk = 0; kk < 16; ++kk) aCur[kk] = aNxt[kk];
  }

  // Merge the two code-halves held by lane nl and lane nl+16 (same x row).
  float oV = __shfl_xor(bestV, 16, 32);
  int   oI = __shfl_xor(bestI, 16, 32);
  if (oV < bestV || (oV == bestV && oI < bestI)) { bestV = oV; bestI = oI; }
  if (h == 0) {
    indices[row] = bestI;
    atomicAdd(&hist[bestI], 1u);
  }
  __syncthreads();
  for (int i = threadIdx.x; i < NUM_K; i += 256) {
    unsigned int c = hist[i];
    if (c) atomicAdd(&counts[i], c);
  }
}

// ---- 4) quantized output + SSE + scatter 1.0 into one-hot ----
__global__ void __launch_bounds__(256)
vq_quantize(const float* __restrict__ x, const float* __restrict__ w,
            const int* __restrict__ indices, float* __restrict__ outq,
            float* __restrict__ enc, float* __restrict__ sse) {
  int gid = blockIdx.x * 256 + threadIdx.x;   // 0 .. N*D-1
  int n = gid >> 6, d = gid & 63;
  int idx = indices[n];
  float q  = w[idx * EMB_D + d];
  float xv = x[gid];
  __builtin_nontemporal_store(q, outq + gid);
  float diff = q - xv;
  float local = diff * diff;
#pragma unroll
  for (int off = 16; off > 0; off >>= 1) local += __shfl_down(local, off, 32);
  if ((threadIdx.x & 31) == 0) atomicAdd(sse, local);
  if (d == 0) enc[(size_t)n * NUM_K + idx] = 1.0f;
}

// ---- 5) scalars: loss, perplexity ----
__global__ void vq_finalize(const unsigned int* __restrict__ counts,
                            const float* __restrict__ sse,
                            float* __restrict__ out_loss,
                            float* __restrict__ out_perp) {
  __shared__ float red[256];
  float acc = 0.f;
  for (int k = (int)threadIdx.x; k < NUM_K; k += 256) {
    float p = (float)counts[k] * (1.0f / (float)NUM_N);
    acc += p * logf(p + 1e-10f);
  }
  red[threadIdx.x] = acc;
  __syncthreads();
  for (int s2 = 128; s2 > 0; s2 >>= 1) {
    if ((int)threadIdx.x < s2) red[threadIdx.x] += red[threadIdx.x + s2];
    __syncthreads();
  }
  if (threadIdx.x == 0) {
    *out_perp = expf(-red[0]);
    *out_loss = 0.25f * (*sse) / (float)((size_t)NUM_N * EMB_D);
  }
}

extern "C" void kernel_launch(void* const* d_in, const int* in_sizes, int n_in,
                              void* d_out, int out_size, void* d_ws, size_t ws_size,
                              hipStream_t stream) {
  const float* x = (const float*)d_in[0];   // [65536, 64]
  const float* w = (const float*)d_in[1];   // [2048, 64]

  char* ws = (char*)d_ws;
  float*        s       = (float*)(ws + WS_S_OFF);
  unsigned int* counts  = (unsigned int*)(ws + WS_CNT_OFF);
  float*        sse     = (float*)(ws + WS_SSE_OFF);
  int*          indices = (int*)(ws + WS_IDX_OFF);

  // Output (f32, concatenated flat): loss[1], quantized[N*D], perplexity[1], encodings[N*K]
  float* out      = (float*)d_out;
  float* out_loss = out;
  float* out_q    = out + 1;
  float* out_perp = out + 1 + (size_t)NUM_N * EMB_D;
  float* out_enc  = out + 2 + (size_t)NUM_N * EMB_D;

  vq_prep    <<<NUM_K / 256, 256, 0, stream>>>(w, s, counts, sse);
  vq_zerofill<<<4096,        256, 0, stream>>>(out_enc);
  vq_argmin  <<<NUM_N / 128, 256, 0, stream>>>(x, w, s, counts, indices);
  vq_quantize<<<(NUM_N * EMB_D) / 256, 256, 0, stream>>>(x, w, indices, out_q, out_enc, sse);
  vq_finalize<<<1,           256, 0, stream>>>(counts, sse, out_loss, out_perp);
}